// DualRelationsPropagation_58686433132591
// MI455X (gfx1250) — compile-verified
//
#include <hip/hip_runtime.h>
#include <hip/hip_bf16.h>

#define H   768
#define NS  128
#define NQ  256
#define NTOT 384
#define RREL 8
#define KC  64

typedef __attribute__((ext_vector_type(8))) float v8f;
typedef __attribute__((ext_vector_type(2))) float v2f;

// ---------------------------------------------------------------------------
// Kernel 1: Wbar[k,h] = mean_r rel_W[r,k,h]   (mean over relations is linear,
// so the 8 per-relation GEMMs collapse into one GEMM against Wbar)
// ---------------------------------------------------------------------------
__global__ void wbar_kernel(const float* __restrict__ relW, float* __restrict__ wbar) {
    int e = blockIdx.x * 256 + threadIdx.x;          // e < H*H
    float s = 0.f;
#pragma unroll
    for (int r = 0; r < RREL; ++r) s += relW[r * (H * H) + e];
    wbar[e] = s * (1.0f / RREL);
}

// ---------------------------------------------------------------------------
// Kernel 2: C[m,n] = sum_h A[m,h] * W[n, colOff+h] + bias[n]
//   A rows come from support (m < NS) or query (m >= NS).
//   One wave computes a 16x16 fp32 tile with V_WMMA_F32_16X16X4_F32 (exact f32).
//   biasMode: 0 = none, 1 = bias[n], 2 = mean_r bias[r*H + n]
// ---------------------------------------------------------------------------
__global__ void gemm_wmma_f32(const float* __restrict__ Asup,
                              const float* __restrict__ Aqry,
                              const float* __restrict__ W, int ldW, int colOff,
                              const float* __restrict__ bias, int biasMode,
                              float* __restrict__ out) {
    const int lane = threadIdx.x & 31;
    const int wv   = threadIdx.x >> 5;
    const int m0   = blockIdx.x * 16;
    const int n0   = (blockIdx.y * 4 + wv) * 16;
    const int l15  = lane & 15;
    const int khalf = (lane >> 4) * 2;               // lanes 0-15: K=0,1 ; 16-31: K=2,3

    const float* Arow = (m0 < NS) ? (Asup + (size_t)(m0 + l15) * H)
                                  : (Aqry + (size_t)(m0 - NS + l15) * H);
    const float* Brow = W + (size_t)(n0 + l15) * ldW + colOff;

    v8f acc = {0.f, 0.f, 0.f, 0.f, 0.f, 0.f, 0.f, 0.f};
#pragma unroll 4
    for (int k = 0; k < H; k += 4) {
        v2f a = *(const v2f*)(Arow + k + khalf);
        v2f b = *(const v2f*)(Brow + k + khalf);
        acc = __builtin_amdgcn_wmma_f32_16x16x4_f32(
            /*neg_a=*/false, a, /*neg_b=*/false, b,
            /*c_mod=*/(short)0, acc, /*reuse_a=*/false, /*reuse_b=*/false);
    }

    const int n = n0 + l15;
    float bv = 0.f;
    if (biasMode == 1) {
        bv = bias[n];
    } else if (biasMode == 2) {
        float s = 0.f;
#pragma unroll
        for (int r = 0; r < RREL; ++r) s += bias[r * H + n];
        bv = s * (1.0f / RREL);
    }

    const int mbase = m0 + ((lane >> 4) ? 8 : 0);    // C/D layout: M = r + 8*(lane>=16)
#pragma unroll
    for (int r8 = 0; r8 < 8; ++r8) {
        out[(size_t)(mbase + r8) * H + n] = acc[r8] + bv;
    }
}

// ---------------------------------------------------------------------------
// Kernel 3: pairwise  m[i,j] = sum_k relu(L[i,k] + R[j,k]) * w2[k] + b2,
// diagonal zeroed.  32x32 output tile / 256-thread block, LDS-staged K-chunks,
// 3 VALU ops per fused element (add, max, fma). blockIdx.z: 0=sim, 1=div.
// ---------------------------------------------------------------------------
__global__ void pair_kernel(const float* __restrict__ Lsim, const float* __restrict__ Rsim,
                            const float* __restrict__ w2sim, const float* __restrict__ b2sim,
                            const float* __restrict__ Ldiv, const float* __restrict__ Rdiv,
                            const float* __restrict__ w2div, const float* __restrict__ b2div,
                            float* __restrict__ outSim, float* __restrict__ outDiv) {
    __shared__ float Ls[32][KC + 1];                 // +1 pad: conflict-free across rows
    __shared__ float Rs[32][KC + 1];
    __shared__ float Ws[KC];

    const float *Lp, *Rp, *w2;
    float b2;
    float* out;
    if (blockIdx.z == 0) { Lp = Lsim; Rp = Rsim; w2 = w2sim; b2 = b2sim[0]; out = outSim; }
    else                 { Lp = Ldiv; Rp = Rdiv; w2 = w2div; b2 = b2div[0]; out = outDiv; }

    const int i0 = blockIdx.y * 32;
    const int j0 = blockIdx.x * 32;
    const int t  = threadIdx.x;
    const int ti = t >> 3;                           // 0..31 row in tile
    const int tj = (t & 7) * 4;                      // 4 consecutive cols

    float a0 = 0.f, a1 = 0.f, a2 = 0.f, a3 = 0.f;

    for (int k0 = 0; k0 < H; k0 += KC) {
        __syncthreads();
#pragma unroll
        for (int idx = t; idx < 32 * KC; idx += 256) {
            int r = idx >> 6;                        // KC == 64
            int c = idx & (KC - 1);
            Ls[r][c] = Lp[(size_t)(i0 + r) * H + k0 + c];
            Rs[r][c] = Rp[(size_t)(j0 + r) * H + k0 + c];
        }
        if (t < KC) Ws[t] = w2[k0 + t];
        __syncthreads();
#pragma unroll 4
        for (int k = 0; k < KC; ++k) {
            float lv = Ls[ti][k];
            float wv = Ws[k];
            a0 += fmaxf(lv + Rs[tj + 0][k], 0.f) * wv;
            a1 += fmaxf(lv + Rs[tj + 1][k], 0.f) * wv;
            a2 += fmaxf(lv + Rs[tj + 2][k], 0.f) * wv;
            a3 += fmaxf(lv + Rs[tj + 3][k], 0.f) * wv;
        }
    }

    const int i = i0 + ti;
    const int j = j0 + tj;
    float v[4] = {a0 + b2, a1 + b2, a2 + b2, a3 + b2};
#pragma unroll
    for (int c = 0; c < 4; ++c) {
        out[(size_t)i * NTOT + j + c] = (i == j + c) ? 0.f : v[c];
    }
}

// ---------------------------------------------------------------------------
extern "C" void kernel_launch(void* const* d_in, const int* in_sizes, int n_in,
                              void* d_out, int out_size, void* d_ws, size_t ws_size,
                              hipStream_t stream) {
    const float* sup    = (const float*)d_in[0];
    // d_in[1] = support_labels (unused by the reference outputs)
    const float* qry    = (const float*)d_in[2];
    const float* simW1  = (const float*)d_in[3];
    const float* simb1  = (const float*)d_in[4];
    const float* simw2  = (const float*)d_in[5];
    const float* simb2  = (const float*)d_in[6];
    const float* divW1  = (const float*)d_in[7];
    const float* divb1  = (const float*)d_in[8];
    const float* divw2  = (const float*)d_in[9];
    const float* divb2  = (const float*)d_in[10];
    const float* relW   = (const float*)d_in[11];
    const float* relb   = (const float*)d_in[12];

    float* out = (float*)d_out;
    float* ws  = (float*)d_ws;

    // workspace layout (floats)
    float* wbar = ws;                                // 768*768      = 589824
    float* Lsim = ws + 589824;                       // 384*768      = 294912
    float* Rsim = Lsim + 294912;
    float* Ldiv = Rsim + 294912;
    float* Rdiv = Ldiv + 294912;

    // output layout (floats): prop(384*768) | sim(384*384) | div(384*384)
    float* outProp = out;
    float* outSim  = out + NTOT * H;                 // 294912
    float* outDiv  = outSim + NTOT * NTOT;           // +147456

    // 1) Wbar = mean_r rel_W[r]
    wbar_kernel<<<(H * H) / 256, 256, 0, stream>>>(relW, wbar);

    // 2) WMMA GEMMs: 24 row-tiles x (12 blocks * 4 waves) col-tiles
    dim3 ggrid(NTOT / 16, (H / 16) / 4);
    dim3 gblk(128);
    // prop = emb @ Wbar.T + mean_r rel_b
    gemm_wmma_f32<<<ggrid, gblk, 0, stream>>>(sup, qry, wbar, H, 0, relb, 2, outProp);
    // sim: L = emb @ W1[:,:H].T + b1 ;  R = emb @ W1[:,H:].T
    gemm_wmma_f32<<<ggrid, gblk, 0, stream>>>(sup, qry, simW1, 2 * H, 0, simb1, 1, Lsim);
    gemm_wmma_f32<<<ggrid, gblk, 0, stream>>>(sup, qry, simW1, 2 * H, H, nullptr, 0, Rsim);
    // div: same structure
    gemm_wmma_f32<<<ggrid, gblk, 0, stream>>>(sup, qry, divW1, 2 * H, 0, divb1, 1, Ldiv);
    gemm_wmma_f32<<<ggrid, gblk, 0, stream>>>(sup, qry, divW1, 2 * H, H, nullptr, 0, Rdiv);

    // 3) pairwise relu-reduction for sim (z=0) and div (z=1)
    dim3 pgrid(NTOT / 32, NTOT / 32, 2);
    pair_kernel<<<pgrid, 256, 0, stream>>>(Lsim, Rsim, simw2, simb2,
                                           Ldiv, Rdiv, divw2, divb2,
                                           outSim, outDiv);
}